// Net_LSTM_2516850835908
// MI455X (gfx1250) — compile-verified
//
#include <hip/hip_runtime.h>

typedef float v2f __attribute__((ext_vector_type(2)));
typedef float v8f __attribute__((ext_vector_type(8)));

#define LSTM_T   16384
#define LSTM_IN  50
#define LSTM_H   128
#define GATES    512    // 4*H

__device__ __forceinline__ float sigmoidf_(float x) {
    return 1.0f / (1.0f + __expf(-x));
}

// ---------------------------------------------------------------------------
// C[m][n] = sum_k A[m*K+k] * W[n*K+k] + b1[n] + b2[n]   (C = A @ W^T + bias)
// K is compile-time. One wave computes a 16x32 output strip (two 16x16 WMMA
// tiles sharing the A fragment -> two independent accumulation chains).
// All loads are unconditional; the K%4 tail masks *values*, not loads.
// Requires: M%16==0, N%32==0, K even (50 and 128 both are).
// ---------------------------------------------------------------------------
template <int K>
__global__ void gemm_abt_bias_wmma(const float* __restrict__ A,
                                   const float* __restrict__ W,
                                   const float* __restrict__ b1,
                                   const float* __restrict__ b2,
                                   float* __restrict__ C,
                                   int M, int N)
{
    const int wavesPerBlock = blockDim.x >> 5;
    const int wave = blockIdx.x * wavesPerBlock + (threadIdx.x >> 5);
    const int lane = threadIdx.x & 31;
    const int npairs = N >> 5;               // 16x32 strips along N
    const int total  = (M >> 4) * npairs;
    if (wave >= total) return;

    const int np = wave % npairs;
    const int mt = wave / npairs;
    const int m0 = mt << 4, n0 = np << 5;
    const int half  = lane >> 4;             // 0: K+{0,1}, 1: K+{2,3}
    const int l     = lane & 15;
    const int kbase = half << 1;

    const float* __restrict__ Arow = A + (size_t)(m0 + l) * K;
    const float* __restrict__ W0r  = W + (size_t)(n0 + l) * K;
    const float* __restrict__ W1r  = W + (size_t)(n0 + 16 + l) * K;

    v8f acc0 = {0.f, 0.f, 0.f, 0.f, 0.f, 0.f, 0.f, 0.f};
    v8f acc1 = {0.f, 0.f, 0.f, 0.f, 0.f, 0.f, 0.f, 0.f};

    constexpr int KFULL = K & ~3;
#pragma unroll
    for (int k0 = 0; k0 < KFULL; k0 += 4) {
        const int ka = k0 + kbase;           // even -> 8B aligned
        const v2f a  = *(const v2f*)(Arow + ka);
        const v2f bA = *(const v2f*)(W0r  + ka);
        const v2f bB = *(const v2f*)(W1r  + ka);
        acc0 = __builtin_amdgcn_wmma_f32_16x16x4_f32(false, a, false, bA,
                                                     (short)0, acc0, false, false);
        acc1 = __builtin_amdgcn_wmma_f32_16x16x4_f32(false, a, false, bB,
                                                     (short)0, acc1, false, false);
    }
    if constexpr (KFULL != K) {
        // Tail chunk (K%4 == 2): clamp address to last valid pair, mask values.
        const int ka = KFULL + kbase;
        const int kc = (ka + 1 < K) ? ka : (K - 2);       // always in-bounds
        const float m0f = (ka     < K) ? 1.0f : 0.0f;
        const float m1f = (ka + 1 < K) ? 1.0f : 0.0f;
        const v2f at  = *(const v2f*)(Arow + kc);
        const v2f btA = *(const v2f*)(W0r  + kc);
        const v2f btB = *(const v2f*)(W1r  + kc);
        v2f a, bA, bB;
        a.x  = at.x  * m0f;  a.y  = at.y  * m1f;
        bA.x = btA.x * m0f;  bA.y = btA.y * m1f;
        bB.x = btB.x * m0f;  bB.y = btB.y * m1f;
        acc0 = __builtin_amdgcn_wmma_f32_16x16x4_f32(false, a, false, bA,
                                                     (short)0, acc0, false, false);
        acc1 = __builtin_amdgcn_wmma_f32_16x16x4_f32(false, a, false, bB,
                                                     (short)0, acc1, false, false);
    }

    const int nA = n0 + l;
    const int nB = n0 + 16 + l;
    const float biasA = b1[nA] + b2[nA];
    const float biasB = b1[nB] + b2[nB];
#pragma unroll
    for (int v = 0; v < 8; ++v) {
        const int m = m0 + v + (half << 3);  // D layout: M = v (+8 for hi half)
        C[(size_t)m * N + nA] = acc0[v] + biasA;
        C[(size_t)m * N + nB] = acc1[v] + biasB;
    }
}

// ---------------------------------------------------------------------------
// Sequential LSTM scan, one persistent workgroup (512 threads = 16 waves).
// Thread j owns gate row j: W_hh[j][0..127] pinned in VGPRs; h broadcast
// from LDS (uniform address -> bank broadcast); threads 0..127 apply the
// pointwise cell update.
// pre: [T, 512] (input contribution + biases), h_out: [T, 128]
// ---------------------------------------------------------------------------
__global__ void __launch_bounds__(512, 1)
lstm_scan(const float* __restrict__ pre,
          const float* __restrict__ W_hh,   // [512,128]
          float* __restrict__ h_out,
          int T)
{
    __shared__ __align__(16) float h_s[LSTM_H];
    __shared__ float g_s[GATES];

    const int j = threadIdx.x;              // 0..511

    float w[LSTM_H];
#pragma unroll
    for (int k = 0; k < LSTM_H; ++k) w[k] = W_hh[j * LSTM_H + k];

    float c = 0.0f;
    if (j < LSTM_H) h_s[j] = 0.0f;
    __syncthreads();

    for (int t = 0; t < T; ++t) {
        float acc = pre[(size_t)t * GATES + j];
        if (t + 1 < T)
            __builtin_prefetch(pre + (size_t)(t + 1) * GATES + j, 0, 1);
#pragma unroll
        for (int k = 0; k < LSTM_H; k += 4) {
            const float4 hv = *(const float4*)&h_s[k];
            acc += w[k]     * hv.x;
            acc += w[k + 1] * hv.y;
            acc += w[k + 2] * hv.z;
            acc += w[k + 3] * hv.w;
        }
        g_s[j] = acc;
        __syncthreads();

        if (j < LSTM_H) {
            const float ig = sigmoidf_(g_s[j]);
            const float fg = sigmoidf_(g_s[j + LSTM_H]);
            const float gg = tanhf(g_s[j + 2 * LSTM_H]);
            const float og = sigmoidf_(g_s[j + 3 * LSTM_H]);
            c = fg * c + ig * gg;
            const float h = og * tanhf(c);
            h_s[j] = h;
            h_out[(size_t)t * LSTM_H + j] = h;
        }
        __syncthreads();
    }
}

// ---------------------------------------------------------------------------
// out[t][o] = sum_k h2[t][k] * fc_W[o][k] + fc_b[o]   (OUT = 5)
// ---------------------------------------------------------------------------
__global__ void fc_head(const float* __restrict__ h2,
                        const float* __restrict__ fc_W,  // [5,128]
                        const float* __restrict__ fc_b,  // [5]
                        float* __restrict__ out,
                        int T)
{
    const int t = blockIdx.x * blockDim.x + threadIdx.x;
    if (t >= T) return;

    float acc[5];
#pragma unroll
    for (int o = 0; o < 5; ++o) acc[o] = fc_b[o];

    const float* __restrict__ hrow = h2 + (size_t)t * LSTM_H;
#pragma unroll 4
    for (int k = 0; k < LSTM_H; ++k) {
        const float hv = hrow[k];
#pragma unroll
        for (int o = 0; o < 5; ++o) acc[o] += hv * fc_W[o * LSTM_H + k];
    }
#pragma unroll
    for (int o = 0; o < 5; ++o) out[(size_t)t * 5 + o] = acc[o];
}

// ---------------------------------------------------------------------------
extern "C" void kernel_launch(void* const* d_in, const int* in_sizes, int n_in,
                              void* d_out, int out_size, void* d_ws, size_t ws_size,
                              hipStream_t stream)
{
    (void)in_sizes; (void)n_in; (void)out_size; (void)ws_size;

    const float* x     = (const float*)d_in[0];   // [16384, 50]
    const float* W_ih1 = (const float*)d_in[1];   // [512, 50]
    const float* W_hh1 = (const float*)d_in[2];   // [512, 128]
    const float* b_ih1 = (const float*)d_in[3];   // [512]
    const float* b_hh1 = (const float*)d_in[4];   // [512]
    const float* W_ih2 = (const float*)d_in[5];   // [512, 128]
    const float* W_hh2 = (const float*)d_in[6];   // [512, 128]
    const float* b_ih2 = (const float*)d_in[7];   // [512]
    const float* b_hh2 = (const float*)d_in[8];   // [512]
    const float* fc_W  = (const float*)d_in[9];   // [5, 128]
    const float* fc_b  = (const float*)d_in[10];  // [5]
    float* out = (float*)d_out;                   // [16384, 5]

    // Workspace: pre (reused by both layers) | h1 | h2   (~48 MB)
    float* pre = (float*)d_ws;                        // 16384*512
    float* h1  = pre + (size_t)LSTM_T * GATES;        // 16384*128
    float* h2  = h1  + (size_t)LSTM_T * LSTM_H;       // 16384*128

    // One wave per 16x32 strip, 8 waves / block.
    const int strips = (LSTM_T / 16) * (GATES / 32);  // 1024 * 16 = 16384 waves
    const int blocks = (strips * 32 + 255) / 256;     // 2048 blocks

    // Layer 1: pre = x @ W_ih1^T + (b_ih1 + b_hh1)
    gemm_abt_bias_wmma<LSTM_IN><<<blocks, 256, 0, stream>>>(
        x, W_ih1, b_ih1, b_hh1, pre, LSTM_T, GATES);
    // Layer 1 recurrence -> h1
    lstm_scan<<<1, 512, 0, stream>>>(pre, W_hh1, h1, LSTM_T);

    // Layer 2: pre = h1 @ W_ih2^T + (b_ih2 + b_hh2)
    gemm_abt_bias_wmma<LSTM_H><<<blocks, 256, 0, stream>>>(
        h1, W_ih2, b_ih2, b_hh2, pre, LSTM_T, GATES);
    // Layer 2 recurrence -> h2
    lstm_scan<<<1, 512, 0, stream>>>(pre, W_hh2, h2, LSTM_T);

    // FC head
    fc_head<<<(LSTM_T + 127) / 128, 128, 0, stream>>>(h2, fc_W, fc_b, out, LSTM_T);
}